// PerfectTreeTraversalTreeImpl_50483045597802
// MI455X (gfx1250) — compile-verified
//
#include <hip/hip_runtime.h>
#include <stdint.h>

#define BATCH 65536
#define FEATS 256
#define TREES 256
#define NCLS  4
#define ROWS  8             // batch rows per tile (8 KB x tile)
#define TPB   256           // 8 waves of 32; one tree per thread
#define NBLK  1024          // persistent blocks; 8 tiles each
#define NTILES (BATCH / ROWS)
#define PK_TOTAL 15872      // 512 + 1024 + 2048 + 4096 + 8192 packed (node,bias) pairs

typedef float    v4f __attribute__((ext_vector_type(4)));
typedef unsigned v2u __attribute__((ext_vector_type(2)));

// Generic (flat) address of a __shared__ object carries the LDS byte offset in
// its low 32 bits (ISA 10.2, LDS aperture).
static __device__ __forceinline__ unsigned lds_off(const void* p) {
    return (unsigned)(uintptr_t)p;
}

// CDNA5 async global->LDS copy (GVS mode), tracked by ASYNCcnt.
static __device__ __forceinline__ void async_load_b128(unsigned lds_addr,
                                                       unsigned byte_off,
                                                       uint64_t base) {
    asm volatile("global_load_async_to_lds_b128 %0, %1, %2"
                 :
                 : "v"(lds_addr), "v"(byte_off), "s"(base)
                 : "memory");
}
static __device__ __forceinline__ void wait_async0() {
    asm volatile("s_wait_asynccnt 0" ::: "memory");
}

// Monotone float -> u32 mapping (order-preserving for all finite values).
static __device__ __forceinline__ unsigned ordered_key(float f) {
    unsigned u = __float_as_uint(f);
    return u ^ ((unsigned)((int)u >> 31) | 0x80000000u);
}

// One-shot packer: interleave (node, bias) per level into d_ws so each
// traversal level is a single b64 gather.
__global__ __launch_bounds__(256) void pack_tables_kernel(
    const int* __restrict__ n1, const float* __restrict__ b1,
    const int* __restrict__ n2, const float* __restrict__ b2,
    const int* __restrict__ n3, const float* __restrict__ b3,
    const int* __restrict__ n4, const float* __restrict__ b4,
    const int* __restrict__ n5, const float* __restrict__ b5,
    v2u* __restrict__ pk)
{
    const int i = blockIdx.x * blockDim.x + threadIdx.x;
    if (i >= PK_TOTAL) return;
    const int* n; const float* b; int off;
    if      (i < 512)  { n = n1; b = b1; off = 0;    }
    else if (i < 1536) { n = n2; b = b2; off = 512;  }
    else if (i < 3584) { n = n3; b = b3; off = 1536; }
    else if (i < 7680) { n = n4; b = b4; off = 3584; }
    else               { n = n5; b = b5; off = 7680; }
    const int j = i - off;
    v2u p;
    p.x = (unsigned)n[j];
    p.y = __float_as_uint(b[j]);
    pk[i] = p;
}

template <bool PACKED>
__global__ __launch_bounds__(TPB) void forest_traverse_kernel(
    const float* __restrict__ x,
    const int*   __restrict__ root_nodes,
    const float* __restrict__ root_biases,
    const float* __restrict__ leaf,          // [TREES*64, NCLS]
    const v2u*   __restrict__ pk,            // packed level tables (if PACKED)
    const int* __restrict__ n1, const float* __restrict__ b1,
    const int* __restrict__ n2, const float* __restrict__ b2,
    const int* __restrict__ n3, const float* __restrict__ b3,
    const int* __restrict__ n4, const float* __restrict__ b4,
    const int* __restrict__ n5, const float* __restrict__ b5,
    int*   __restrict__ amax,                // [BATCH, NCLS]
    float* __restrict__ out)                 // [BATCH, TREES, NCLS]
{
    __shared__ float xs[2][ROWS][FEATS];                 // 16 KB, double-buffered
    __shared__ unsigned long long slots[ROWS * NCLS];    // 32 argmax slots (all 64 banks)

    const int tid = threadIdx.x;                         // == tree index

    // Root split parameters are per-tree constants.
    const int   rf = root_nodes[tid];
    const float rb = root_biases[tid];

    const unsigned lds0   = lds_off(&xs[0][0][0]);
    const unsigned lds1   = lds_off(&xs[1][0][0]);
    const uint64_t xbase  = (uint64_t)(uintptr_t)x;
    const unsigned tbytes = ROWS * FEATS * 4;            // 8192 B / tile

    // Warm WGP$/L2 with the gather tables.
    {
        const char* plf = (const char*)leaf;
        for (int i = tid * 64; i < TREES * 64 * NCLS * 4; i += TPB * 64)
            __builtin_prefetch(plf + i, 0, 3);
        if (PACKED) {
            const char* ppk = (const char*)pk;
            for (int i = tid * 64; i < PK_TOTAL * 8; i += TPB * 64)
                __builtin_prefetch(ppk + i, 0, 3);
        } else {
            const char* p5n = (const char*)n5;
            const char* p5b = (const char*)b5;
            for (int i = tid * 64; i < TREES * 32 * 4; i += TPB * 64) {
                __builtin_prefetch(p5n + i, 0, 3);
                __builtin_prefetch(p5b + i, 0, 3);
            }
        }
    }

    int tile = blockIdx.x;
    {   // Prologue: async fill of buffer 0 (512 b128 chunks, 2 per thread).
        unsigned goff = (unsigned)tile * tbytes;
        async_load_b128(lds0 + (unsigned)tid * 16u,
                        goff + (unsigned)tid * 16u, xbase);
        async_load_b128(lds0 + (unsigned)(tid + TPB) * 16u,
                        goff + (unsigned)(tid + TPB) * 16u, xbase);
    }

    int buf = 0;
    for (; tile < NTILES; tile += gridDim.x) {
        wait_async0();          // my wave's async fills done
        __syncthreads();        // all fills visible; prev finalize done

        if (tid < ROWS * NCLS) slots[tid] = 0ull;

        // Prefetch next tile into the other buffer (overlaps with compute).
        const int ntile = tile + (int)gridDim.x;
        if (ntile < NTILES) {
            const unsigned nlds = (buf == 0) ? lds1 : lds0;
            unsigned goff = (unsigned)ntile * tbytes;
            async_load_b128(nlds + (unsigned)tid * 16u,
                            goff + (unsigned)tid * 16u, xbase);
            async_load_b128(nlds + (unsigned)(tid + TPB) * 16u,
                            goff + (unsigned)(tid + TPB) * 16u, xbase);
        }
        __syncthreads();        // slot init visible before atomics

        #pragma unroll
        for (int r = 0; r < ROWS; ++r) {
            const float* xr = &xs[buf][r][0];
            const int b = tile * ROWS + r;

            // Root split: prev in [0, 2T)
            int prev = 2 * tid + ((xr[rf] >= rb) ? 1 : 0);

            // Levels 1..5: one b64 pair-gather per level (PACKED) or two b32.
            #define TREE_STEP(OFF, NP, BP)                                     \
                {                                                              \
                    int f; float bias;                                         \
                    if (PACKED) {                                              \
                        v2u p = pk[(OFF) + prev];                              \
                        f = (int)p.x; bias = __uint_as_float(p.y);             \
                    } else {                                                   \
                        f = NP[prev]; bias = BP[prev];                         \
                    }                                                          \
                    prev = 2 * prev + ((xr[f] >= bias) ? 1 : 0);               \
                }
            TREE_STEP(0,    n1, b1)
            TREE_STEP(512,  n2, b2)
            TREE_STEP(1536, n3, b3)
            TREE_STEP(3584, n4, b4)
            TREE_STEP(7680, n5, b5)
            #undef TREE_STEP

            // Leaf row gather: 16 B aligned, L2-resident.
            const v4f lv = ((const v4f*)leaf)[prev];

            // Coalesced non-temporal store of out[b][t][0..3].
            __builtin_nontemporal_store(lv, ((v4f*)out) + (size_t)b * TREES + tid);

            // argmax over trees via packed-key LDS atomic max:
            //   key = ordered(val)<<32 | ~tree  -> larger key == larger val,
            //   and on value ties the smaller tree index wins (first occurrence).
            const unsigned lowk = (unsigned)(~tid);
            unsigned long long* sl = &slots[r * NCLS];
            atomicMax(&sl[0], ((unsigned long long)ordered_key(lv.x) << 32) | lowk);
            atomicMax(&sl[1], ((unsigned long long)ordered_key(lv.y) << 32) | lowk);
            atomicMax(&sl[2], ((unsigned long long)ordered_key(lv.z) << 32) | lowk);
            atomicMax(&sl[3], ((unsigned long long)ordered_key(lv.w) << 32) | lowk);
        }
        __syncthreads();        // all atomics done (DScnt drained at barrier)

        if (tid < ROWS * NCLS) {           // tid = r*NCLS + c
            const unsigned long long k = slots[tid];
            const int bi = (int)(~(unsigned)k);      // recover tree index
            const int r = tid >> 2, c = tid & 3;
            amax[(size_t)(tile * ROWS + r) * NCLS + c] = bi;
        }
        buf ^= 1;
        // Next iteration's wait+sync protects slots and the x buffers.
    }
}

extern "C" void kernel_launch(void* const* d_in, const int* in_sizes, int n_in,
                              void* d_out, int out_size, void* d_ws, size_t ws_size,
                              hipStream_t stream) {
    const float* x    = (const float*)d_in[0];
    const int*   rn   = (const int*)d_in[1];
    const float* rb   = (const float*)d_in[2];
    const float* leaf = (const float*)d_in[3];
    const int*   n1   = (const int*)d_in[4];   const float* b1 = (const float*)d_in[5];
    const int*   n2   = (const int*)d_in[6];   const float* b2 = (const float*)d_in[7];
    const int*   n3   = (const int*)d_in[8];   const float* b3 = (const float*)d_in[9];
    const int*   n4   = (const int*)d_in[10];  const float* b4 = (const float*)d_in[11];
    const int*   n5   = (const int*)d_in[12];  const float* b5 = (const float*)d_in[13];

    // Tuple output, flat-concatenated: argmax [B, C] (int32 bits), then
    // out [B, T, C] (f32). Float offset B*C -> 1 MB, 16 B aligned.
    int*   amax = (int*)d_out;
    float* out  = (float*)d_out + (size_t)BATCH * NCLS;

    const bool packed = ws_size >= (size_t)PK_TOTAL * 8u;
    if (packed) {
        v2u* pk = (v2u*)d_ws;
        pack_tables_kernel<<<(PK_TOTAL + 255) / 256, 256, 0, stream>>>(
            n1, b1, n2, b2, n3, b3, n4, b4, n5, b5, pk);
        forest_traverse_kernel<true><<<NBLK, TPB, 0, stream>>>(
            x, rn, rb, leaf, pk,
            n1, b1, n2, b2, n3, b3, n4, b4, n5, b5, amax, out);
    } else {
        forest_traverse_kernel<false><<<NBLK, TPB, 0, stream>>>(
            x, rn, rb, leaf, (const v2u*)nullptr,
            n1, b1, n2, b2, n3, b3, n4, b4, n5, b5, amax, out);
    }
}